// CompGCNLayer_69475390980298
// MI455X (gfx1250) — compile-verified
//
#include <hip/hip_runtime.h>
#include <cstdint>
#include <cstddef>

#define DIM 128
#define LDS_STRIDE 132   // 128 + 4 dword pad -> b64 frag reads hit all 64 banks

typedef __attribute__((ext_vector_type(2))) float v2f;
typedef __attribute__((ext_vector_type(8))) float v8f;

__device__ __forceinline__ v8f wmma4(v2f a, v2f b, v8f c) {
  // D = A(16x4 f32) x B(4x16 f32) + C(16x16 f32)
  return __builtin_amdgcn_wmma_f32_16x16x4_f32(false, a, false, b, (short)0, c,
                                               false, false);
}

// ---------------- sigmoid gate table: sig = 1/(1+exp(-rel)) ----------------
__global__ void sigmoid_kernel(const float* __restrict__ rel,
                               float* __restrict__ sig, int n) {
  int i = blockIdx.x * blockDim.x + threadIdx.x;
  if (i < n) {
    float x = rel[i];
    sig[i] = 1.0f / (1.0f + __expf(-x));
  }
}

// ---------------- zero the aggregation buffer ----------------
__global__ void zero_kernel(float4* __restrict__ p, int n4) {
  int i = blockIdx.x * blockDim.x + threadIdx.x;
  if (i < n4) p[i] = make_float4(0.f, 0.f, 0.f, 0.f);
}

// ---------------- gather -> gate -> scatter-add ----------------
// one wave per edge; lane handles 4 dims (float4); hw fp32 atomics in L2
__global__ void scatter_kernel(const float* __restrict__ ent,
                               const float* __restrict__ sig,
                               const int* __restrict__ src,
                               const int* __restrict__ dst,
                               const int* __restrict__ etype,
                               float* __restrict__ agg, int E) {
  int wave = threadIdx.x >> 5;
  int lane = threadIdx.x & 31;
  int e = blockIdx.x * (blockDim.x >> 5) + wave;
  if (e >= E) return;
  int s = src[e];
  int d = dst[e];
  int t = etype[e];
  float4 a = ((const float4*)(ent + (size_t)s * DIM))[lane];
  float4 g = ((const float4*)(sig + (size_t)t * DIM))[lane];
  float* o = agg + (size_t)d * DIM + lane * 4;
  __hip_atomic_fetch_add(o + 0, a.x * g.x, __ATOMIC_RELAXED, __HIP_MEMORY_SCOPE_AGENT);
  __hip_atomic_fetch_add(o + 1, a.y * g.y, __ATOMIC_RELAXED, __HIP_MEMORY_SCOPE_AGENT);
  __hip_atomic_fetch_add(o + 2, a.z * g.z, __ATOMIC_RELAXED, __HIP_MEMORY_SCOPE_AGENT);
  __hip_atomic_fetch_add(o + 3, a.w * g.w, __ATOMIC_RELAXED, __HIP_MEMORY_SCOPE_AGENT);
}

// ---------------- dual WMMA GEMM: out = act(A1*W1^T (+ A2*W2^T) + b) -------
// block = 256 thr = 8 waves; block -> one 32-row M slab staged in LDS,
// wave w -> N tile [16w,16w+16), each wave holds TWO 16x16 accumulators
// (rows m0..m0+15 and m0+16..m0+31) sharing one B fragment per K-step.
// K = N = DIM = 128; weights row-major [n,k] == B with K contiguous.
__global__ void gemm_wmma_kernel(const float* __restrict__ A1,
                                 const float* __restrict__ W1,
                                 const float* __restrict__ bias1,
                                 const float* __restrict__ A2,   // nullable
                                 const float* __restrict__ W2,
                                 const float* __restrict__ bias2,
                                 float* __restrict__ out,
                                 int M, int do_relu) {
  __shared__ float tA1[32 * LDS_STRIDE];
  __shared__ float tA2[32 * LDS_STRIDE];

  const int wave = threadIdx.x >> 5;
  const int lane = threadIdx.x & 31;
  const int half = lane >> 4;   // selects K pair within a 4-wide K step
  const int lm   = lane & 15;   // M idx (A frags) / N idx (B frags)
  const int m0 = blockIdx.x * 32;
  const int n0 = wave * 16;
  const bool dual = (A2 != nullptr);

  // ---- stage A slab(s): 32 rows x 128 f32, coalesced float4, row-clamped
  for (int p = threadIdx.x; p < 32 * 32; p += 256) {  // p in float4 units
    int row = p >> 5, c4 = p & 31;
    int gr = m0 + row;
    if (gr > M - 1) gr = M - 1;
    float4 v = ((const float4*)(A1 + (size_t)gr * DIM))[c4];
    *(float4*)&tA1[row * LDS_STRIDE + c4 * 4] = v;
    if (dual) {
      float4 u = ((const float4*)(A2 + (size_t)gr * DIM))[c4];
      *(float4*)&tA2[row * LDS_STRIDE + c4 * 4] = u;
    }
  }
  __syncthreads();

  const float* w1 = W1 + (size_t)(n0 + lm) * DIM + 2 * half;
  const float* a_lo = &tA1[(size_t)lm * LDS_STRIDE + 2 * half];
  const float* a_hi = a_lo + 16 * LDS_STRIDE;

  v8f acc0 = {};
  v8f acc1 = {};
#pragma unroll
  for (int k = 0; k < DIM; k += 4) {
    v2f bv = *(const v2f*)(w1 + k);        // B[k+2h.., n] = W1[n, k+2h..]
    v2f a0 = *(const v2f*)(a_lo + k);      // rows m0..m0+15
    v2f a1 = *(const v2f*)(a_hi + k);      // rows m0+16..m0+31
    acc0 = wmma4(a0, bv, acc0);
    acc1 = wmma4(a1, bv, acc1);
  }

  if (dual) {
    const float* w2 = W2 + (size_t)(n0 + lm) * DIM + 2 * half;
    const float* c_lo = &tA2[(size_t)lm * LDS_STRIDE + 2 * half];
    const float* c_hi = c_lo + 16 * LDS_STRIDE;
#pragma unroll
    for (int k = 0; k < DIM; k += 4) {
      v2f bv = *(const v2f*)(w2 + k);
      v2f a0 = *(const v2f*)(c_lo + k);
      v2f a1 = *(const v2f*)(c_hi + k);
      acc0 = wmma4(a0, bv, acc0);
      acc1 = wmma4(a1, bv, acc1);
    }
  }

  float bsum = bias1[n0 + lm] + (dual ? bias2[n0 + lm] : 0.0f);

#pragma unroll
  for (int r = 0; r < 8; ++r) {
    // C/D layout: vgpr r, lanes 0-15 -> m=r, lanes 16-31 -> m=8+r, n=lm
    int mA = m0 + r + 8 * half;
    if (mA < M) {
      float v = acc0[r] + bsum;
      if (do_relu) v = fmaxf(v, 0.0f);
      out[(size_t)mA * DIM + n0 + lm] = v;
    }
    int mB = mA + 16;
    if (mB < M) {
      float v = acc1[r] + bsum;
      if (do_relu) v = fmaxf(v, 0.0f);
      out[(size_t)mB * DIM + n0 + lm] = v;
    }
  }
}

extern "C" void kernel_launch(void* const* d_in, const int* in_sizes, int n_in,
                              void* d_out, int out_size, void* d_ws, size_t ws_size,
                              hipStream_t stream) {
  const float* ent        = (const float*)d_in[0];
  const float* rel        = (const float*)d_in[1];
  const int*   edge_index = (const int*)d_in[2];
  const int*   edge_type  = (const int*)d_in[3];
  const float* W_self     = (const float*)d_in[4];
  const float* b_self     = (const float*)d_in[5];
  const float* W_nei      = (const float*)d_in[6];
  const float* b_nei      = (const float*)d_in[7];
  const float* W_rel      = (const float*)d_in[8];
  const float* b_rel      = (const float*)d_in[9];

  const int N = in_sizes[0] / DIM;   // 100000 nodes
  const int R = in_sizes[1] / DIM;   // 500 relations
  const int E = in_sizes[3];         // 600000 edges

  // workspace layout: [agg: N*DIM f32][sig_rel: R*DIM f32]
  float* agg = (float*)d_ws;
  float* sig = (float*)((char*)d_ws + (size_t)N * DIM * sizeof(float));

  float* out_ent = (float*)d_out;
  float* out_rel = out_ent + (size_t)N * DIM;

  const int* src = edge_index;
  const int* dst = edge_index + E;

  // 1) sigmoid gate table (256 KB, L2-resident)
  {
    int n = R * DIM;
    sigmoid_kernel<<<(n + 255) / 256, 256, 0, stream>>>(rel, sig, n);
  }
  // 2) zero agg
  {
    int n4 = (N * DIM) / 4;
    zero_kernel<<<(n4 + 255) / 256, 256, 0, stream>>>((float4*)agg, n4);
  }
  // 3) gather/gate/scatter-add
  {
    int epb = 256 / 32;
    scatter_kernel<<<(E + epb - 1) / epb, 256, 0, stream>>>(
        ent, sig, src, dst, edge_type, agg, E);
  }
  // 4) out_ent = relu(ent@W_self^T + agg@W_nei^T + b_self + b_nei)
  {
    int mblocks = (N + 31) / 32;   // 3125
    gemm_wmma_kernel<<<mblocks, 256, 0, stream>>>(ent, W_self, b_self,
                                                  agg, W_nei, b_nei,
                                                  out_ent, N, 1);
  }
  // 5) out_rel = rel@W_rel^T + b_rel
  {
    int mblocks = (R + 31) / 32;   // 16 (ragged tail clamped/guarded)
    gemm_wmma_kernel<<<mblocks, 256, 0, stream>>>(rel, W_rel, b_rel,
                                                  nullptr, nullptr, nullptr,
                                                  out_rel, R, 0);
  }
}